// OptimalTransport_85693187490377
// MI455X (gfx1250) — compile-verified
//
#include <hip/hip_runtime.h>

#ifndef __has_builtin
#define __has_builtin(x) 0
#endif

#if defined(__AMDGCN__) && \
    __has_builtin(__builtin_amdgcn_global_load_async_to_lds_b128) && \
    __has_builtin(__builtin_amdgcn_s_wait_asynccnt)
#define USE_ASYNC_LDS 1
#else
#define USE_ASYNC_LDS 0
#endif

__device__ __forceinline__ float fexp2(float x) {
#if defined(__AMDGCN__) && __has_builtin(__builtin_amdgcn_exp2f)
  return __builtin_amdgcn_exp2f(x);   // v_exp_f32 (base-2)
#else
  return exp2f(x);
#endif
}
__device__ __forceinline__ float flog2(float x) {
#if defined(__AMDGCN__) && __has_builtin(__builtin_amdgcn_logf)
  return __builtin_amdgcn_logf(x);    // v_log_f32 (base-2)
#else
  return log2f(x);
#endif
}

typedef float v2f __attribute__((ext_vector_type(2)));
typedef float v8f __attribute__((ext_vector_type(8)));
// GCC-style vector to match the async-to-LDS builtin's parameter type
typedef int v4i __attribute__((vector_size(16)));
typedef __attribute__((address_space(1))) v4i gv4i;   // global (printed __device__)
typedef __attribute__((address_space(3))) v4i lv4i;   // LDS

namespace {
constexpr int kThreads    = 512;   // 16 wave32
constexpr int kM          = 455;   // rows (m+1)
constexpr int kN          = 100;   // cols (num queries)
constexpr int kMp         = 456;   // rows padded to multiple of 4 (1 dummy row)
constexpr int kNp         = 128;   // cols padded to 8 groups of 16
constexpr int kHalfRows   = 228;   // kMp / 2
constexpr int kHalfChunks = 57;    // kHalfRows / 4 (WMMA K-chunks, v-pass)
constexpr int kRowGroups  = 29;    // ceil(455 / 16) (u-pass)
constexpr int kColChunks  = 25;    // kN / 4 (WMMA K-chunks, u-pass)
constexpr int kIters      = 50;
// work in base-2 log domain: scale = log2(e)/eps, eps = 0.1
constexpr float kScale    = 14.426950408889634f;
constexpr float kUnscale  = 0.06931471805599453f;  // eps * ln(2) = 1/kScale
constexpr float kNegBig   = -1.0e30f;

struct __align__(16) Smem {
  float Cs[kMp * kN];        // C * kScale, row-major pitch kN; row 455 = 0
  float us[kMp];             // u * kScale; us[455] = -big (pad row -> exp = 0)
  float vs[kNp];             // v * kScale
  float lmu2[kMp];           // log2(mu + 1e-8)
  float lnu2[kNp];           // log2(nu + 1e-8)
  float colM[2 * kNp];       // per-half column max
  float colS[2 * kNp];       // per-half column sum-of-exp2
  float rowM[kRowGroups * 16];
  float rowS[kRowGroups * 16];
  float red[kThreads];
};
}  // namespace

__global__ __launch_bounds__(kThreads, 1)
void sinkhorn_ot_kernel(const float* __restrict__ mu,
                        const float* __restrict__ nu,
                        const float* __restrict__ C,
                        float* __restrict__ out) {
  __shared__ Smem sm;
  const int tid  = threadIdx.x;
  const int lane = tid & 31;
  const int wave = tid >> 5;
  const int hi   = lane >> 4;     // which half of the wave (K split)
  const int nl   = lane & 15;
  const int b    = blockIdx.x;
  const float* __restrict__ Cb = C + (size_t)b * (kM * kN);

  // ---------------- stage C[b] (182 KB) into LDS -------------------------
#if USE_ASYNC_LDS
  {
    const char* gbase = (const char*)Cb;
    char* lbase = (char*)&sm.Cs[0];
    constexpr int nchunk = (kM * kN * 4) / 16;  // 11375 b128 chunks
    for (int c = tid; c < nchunk; c += kThreads) {
      __builtin_amdgcn_global_load_async_to_lds_b128(
          (gv4i*)(gbase + (size_t)c * 16),
          (lv4i*)(lbase + (size_t)c * 16),
          0, 0);
    }
  }
#else
  for (int idx = tid; idx < kM * kN; idx += kThreads) sm.Cs[idx] = Cb[idx];
#endif

  // independent init, overlapped with the async copy
  for (int i = tid; i < kM; i += kThreads) {
    sm.us[i]   = kScale;  // u0 = 1  ->  us = 1 * log2(e)/eps
    sm.lmu2[i] = flog2(mu[(size_t)b * kM + i] + 1e-8f);
  }
  if (tid < kN) sm.lnu2[tid] = flog2(nu[(size_t)b * kN + tid] + 1e-8f);
  if (tid < kN) sm.Cs[kM * kN + tid] = 0.0f;  // zero the pad row 455
  if (tid == 0) sm.us[kM] = kNegBig;          // pad row contributes exp2(-big)=0

#if USE_ASYNC_LDS
  __builtin_amdgcn_s_wait_asynccnt(0);
#endif
  __syncthreads();
  // pre-scale C by log2(e)/eps in place: every later element op = sub + exp2
  for (int idx = tid; idx < kM * kN; idx += kThreads) sm.Cs[idx] *= kScale;
  __syncthreads();

  // v-pass wave roles: 8 column groups x 2 row halves = 16 waves
  const int vg = wave & 7;
  const int vh = wave >> 3;
  const int vj = vg * 16 + nl;                 // 0..127 (>=100 are dummies)
  const int vjc = vj < kN ? vj : kN - 1;       // clamp (results discarded)
  const int kbase0 = vh * kHalfRows;
  const v2f ones2 = {1.0f, 1.0f};

  for (int it = 0; it < kIters; ++it) {
    // ======== v update: vs_j = lnu2_j - log2sumexp2_i(us_i - Cs_ij) ========
    {
      float m = kNegBig;
      for (int c = 0; c < kHalfChunks; ++c) {
        const int r0 = kbase0 + c * 4 + 2 * hi;
        const float t0 = sm.us[r0]     - sm.Cs[r0 * kN + vjc];
        const float t1 = sm.us[r0 + 1] - sm.Cs[(r0 + 1) * kN + vjc];
        m = fmaxf(m, fmaxf(t0, t1));
      }
      m = fmaxf(m, __shfl_xor(m, 16, 32));  // merge K-halves: full half-max
      // sum pass: D += ones(16x4) x E(4x16); every D row = column sums
      v8f acc = {0.f, 0.f, 0.f, 0.f, 0.f, 0.f, 0.f, 0.f};
      for (int c = 0; c < kHalfChunks; ++c) {
        const int r0 = kbase0 + c * 4 + 2 * hi;
        const float t0 = sm.us[r0]     - sm.Cs[r0 * kN + vjc];
        const float t1 = sm.us[r0 + 1] - sm.Cs[(r0 + 1) * kN + vjc];
        v2f bv = {fexp2(t0 - m), fexp2(t1 - m)};
        acc = __builtin_amdgcn_wmma_f32_16x16x4_f32(false, ones2, false, bv,
                                                    (short)0, acc, false, false);
      }
      if (hi == 0) {
        sm.colM[vh * kNp + vj] = m;
        sm.colS[vh * kNp + vj] = acc[0];
      }
    }
    __syncthreads();
    if (tid < kN) {  // merge the two row-halves, finish the LSE
      const float M0 = sm.colM[tid],       Mh = sm.colM[kNp + tid];
      const float S0 = sm.colS[tid],       Sh = sm.colS[kNp + tid];
      const float M  = fmaxf(M0, Mh);
      const float S  = S0 * fexp2(M0 - M) + Sh * fexp2(Mh - M);
      sm.vs[tid] = sm.lnu2[tid] - (M + flog2(S));
    }
    __syncthreads();

    // ======== u update: us_i = lmu2_i - log2sumexp2_j(vs_j - Cs_ij) ========
    for (int rg = wave; rg < kRowGroups; rg += 16) {
      const int m0  = rg * 16;
      const int row = m0 + nl;
      const int rc  = row < kM ? row : kM - 1;  // clamp (results discarded)
      float m = kNegBig;
      for (int c = 0; c < kColChunks; ++c) {
        const int k0 = c * 4 + 2 * hi;
        const float t0 = sm.vs[k0]     - sm.Cs[rc * kN + k0];
        const float t1 = sm.vs[k0 + 1] - sm.Cs[rc * kN + k0 + 1];
        m = fmaxf(m, fmaxf(t0, t1));
      }
      m = fmaxf(m, __shfl_xor(m, 16, 32));  // full row max
      if (hi == 0) sm.rowM[row] = m;
      // sum pass: D += E(16x4) x ones(4x16); every D column = row sums
      v8f acc = {0.f, 0.f, 0.f, 0.f, 0.f, 0.f, 0.f, 0.f};
      for (int c = 0; c < kColChunks; ++c) {
        const int k0 = c * 4 + 2 * hi;
        const float t0 = sm.vs[k0]     - sm.Cs[rc * kN + k0];
        const float t1 = sm.vs[k0 + 1] - sm.Cs[rc * kN + k0 + 1];
        v2f av = {fexp2(t0 - m), fexp2(t1 - m)};
        acc = __builtin_amdgcn_wmma_f32_16x16x4_f32(false, av, false, ones2,
                                                    (short)0, acc, false, false);
      }
      if (nl == 0) {  // lane 0 -> rows m0..m0+7, lane 16 -> rows m0+8..m0+15
        #pragma unroll
        for (int r = 0; r < 8; ++r) sm.rowS[m0 + 8 * hi + r] = acc[r];
      }
    }
    __syncthreads();
    if (tid < kM)
      sm.us[tid] = sm.lmu2[tid] - (sm.rowM[tid] + flog2(sm.rowS[tid]));
    __syncthreads();
  }

  // ======== transport plan pi = exp2(us + vs - Cs) and cost ========
  float accum = 0.0f;
  float* __restrict__ piOut = out + gridDim.x + (size_t)b * (kM * kN);
  for (int idx = tid; idx < kM * kN; idx += kThreads) {
    const int i = idx / kN;
    const int j = idx - i * kN;
    const float cs = sm.Cs[idx];
    const float p  = fexp2(sm.us[i] + sm.vs[j] - cs);
    piOut[idx] = p;
    accum += p * cs;            // sum(pi * Cs); Cs = C * kScale
  }
  sm.red[tid] = accum;
  __syncthreads();
  for (int s = kThreads / 2; s > 0; s >>= 1) {
    if (tid < s) sm.red[tid] += sm.red[tid + s];
    __syncthreads();
  }
  if (tid == 0) out[b] = sm.red[0] * kUnscale;  // recover sum(pi * C)
}

extern "C" void kernel_launch(void* const* d_in, const int* in_sizes, int n_in,
                              void* d_out, int out_size, void* d_ws, size_t ws_size,
                              hipStream_t stream) {
  const float* mu = (const float*)d_in[0];
  const float* nu = (const float*)d_in[1];
  const float* C  = (const float*)d_in[2];
  float* out = (float*)d_out;
  const int B = in_sizes[1] / kN;  // nu is [B, 100]
  sinkhorn_ot_kernel<<<dim3(B), dim3(kThreads), 0, stream>>>(mu, nu, C, out);
  (void)n_in; (void)out_size; (void)d_ws; (void)ws_size;
}